// MM_Model_81114752352534
// MI455X (gfx1250) — compile-verified
//
#include <hip/hip_runtime.h>
#include <hip/hip_bf16.h>

typedef float v2f __attribute__((ext_vector_type(2)));
typedef float v4f __attribute__((ext_vector_type(4)));
typedef float v8f __attribute__((ext_vector_type(8)));

#define NOUT 64
#define KC   64
#define KPAD (KC + 4)   // 68-float row: 16B-aligned rows, conflict-free b64 reads

// ---------------------------------------------------------------------------
// GEMM: Y[M x 64] = X[M x K] @ W[64 x K]^T + bias, fp32 WMMA 16x16x4.
// Block = 256 threads = 8 waves; each wave owns a 16-row strip and all 4
// 16-wide N tiles (4 x v8f accumulators). W panel staged in LDS per K-chunk
// in [n][k] order so every B fragment is one contiguous ds_load_b64.
// ---------------------------------------------------------------------------
__global__ __launch_bounds__(256) void gemm_f32_wmma(
    const float* __restrict__ X, const float* __restrict__ W,
    const float* __restrict__ bias, float* __restrict__ Y, int M, int K)
{
  __shared__ float Bs[NOUT][KPAD];     // Bs[n][k] = W[n][kc+k]
  const int tid   = threadIdx.x;
  const int wave  = tid >> 5;
  const int lane  = tid & 31;
  const int lhalf = lane >> 4;         // 0: K pair {0,1}, 1: K pair {2,3}
  const int lmod  = lane & 15;
  const int rowBase = blockIdx.x * 128 + wave * 16;

  v8f acc[4];
  #pragma unroll
  for (int t = 0; t < 4; ++t) acc[t] = (v8f){0,0,0,0,0,0,0,0};

  int arow = rowBase + lmod;
  if (arow >= M) arow = M - 1;         // clamp loads; stores guarded below
  const float* Xrow = X + (long long)arow * K;

  for (int kc = 0; kc < K; kc += KC) {
    // Stage W chunk: coalesced float4 global reads, one aligned b128 LDS store.
    #pragma unroll
    for (int i = 0; i < 4; ++i) {
      int flat = tid + i * 256;        // 1024 float4 granules
      int n    = flat >> 4;            // 0..63
      int k4   = (flat & 15) << 2;     // 0,4,...,60
      v4f w4 = *(const v4f*)(W + (long long)n * K + kc + k4);
      *(v4f*)(&Bs[n][k4]) = w4;
    }
    __syncthreads();
    if (kc + KC < K) __builtin_prefetch(Xrow + kc + KC, 0, 0);

    #pragma unroll 4
    for (int k = 0; k < KC; k += 4) {
      // A fragment (ISA 16x4 f32 layout): lanes 0-15 -> K=k,k+1 ; 16-31 -> K=k+2,k+3
      v2f a = *(const v2f*)(Xrow + kc + k + 2 * lhalf);
      #pragma unroll
      for (int t = 0; t < 4; ++t) {
        int n = t * 16 + lmod;
        v2f b = *(const v2f*)(&Bs[n][k + 2 * lhalf]);   // contiguous b64
        acc[t] = __builtin_amdgcn_wmma_f32_16x16x4_f32(
            false, a, false, b, (short)0, acc[t], false, false);
      }
    }
    __syncthreads();
  }

  // D layout: acc[t][r] = element (m = rowBase + r + 8*lhalf, n = t*16 + lmod)
  #pragma unroll
  for (int t = 0; t < 4; ++t) {
    int n = t * 16 + lmod;
    float bv = bias[n];
    #pragma unroll
    for (int r = 0; r < 8; ++r) {
      int m = rowBase + r + 8 * lhalf;
      if (m < M) Y[(long long)m * 64 + n] = acc[t][r] + bv;
    }
  }
}

// ---------------------------------------------------------------------------
// SpMM scatter-add: Y[dst[e]] += vals[e] * X[src[e]], one thread per (e, dim)
// ---------------------------------------------------------------------------
__global__ __launch_bounds__(256) void spmm_atomic(
    const int* __restrict__ dst, const int* __restrict__ src,
    const float* __restrict__ vals, const float* __restrict__ X,
    float* __restrict__ Y, int nnz)
{
  unsigned t = blockIdx.x * blockDim.x + threadIdx.x;
  unsigned e = t >> 6;
  unsigned d = t & 63;
  if (e < (unsigned)nnz) {
    float v = vals[e];
    long long r = dst[e], c = src[e];
    atomicAdd(Y + r * 64 + d, v * X[c * 64 + d]);
  }
}

__global__ void zero_f32(float* __restrict__ p, long long n)
{
  long long i = (long long)blockIdx.x * blockDim.x + threadIdx.x;
  if (i < n) p[i] = 0.f;
}

__global__ void fill_iota(float* __restrict__ p, int n)
{
  int i = blockIdx.x * blockDim.x + threadIdx.x;
  if (i < n) p[i] = (float)i;
}

__device__ inline float waveSum(float v) {
  #pragma unroll
  for (int off = 16; off >= 1; off >>= 1) v += __shfl_xor(v, off, 32);
  return v;
}
__device__ inline float waveMax(float v) {
  #pragma unroll
  for (int off = 16; off >= 1; off >>= 1) v = fmaxf(v, __shfl_xor(v, off, 32));
  return v;
}

// Row-wise softmax over 64 dims, one wave32 per row (2 dims per lane)
__global__ __launch_bounds__(256) void softmax64(float* __restrict__ X, int M)
{
  int wave = (int)((blockIdx.x * blockDim.x + threadIdx.x) >> 5);
  int lane = threadIdx.x & 31;
  if (wave >= M) return;
  float* row = X + (long long)wave * 64;
  float x0 = row[lane * 2], x1 = row[lane * 2 + 1];
  float m = waveMax(fmaxf(x0, x1));
  float e0 = __expf(x0 - m), e1 = __expf(x1 - m);
  float s = waveSum(e0 + e1);
  float inv = 1.f / s;
  row[lane * 2]     = e0 * inv;
  row[lane * 2 + 1] = e1 * inv;
}

// out = (e0+e1+e2)/3 + cM*l2n(nA) + cM*l2n(nB) + cC*l2n(nC) + cD*l2n(nD)
__global__ __launch_bounds__(256) void finalize_rows(
    const float* __restrict__ e0, const float* __restrict__ e1,
    const float* __restrict__ e2,
    const float* __restrict__ nA, const float* __restrict__ nB,
    const float* __restrict__ nC, const float* __restrict__ nD,
    float* __restrict__ out, int M, float cM, float cC, float cD)
{
  int wave = (int)((blockIdx.x * blockDim.x + threadIdx.x) >> 5);
  int lane = threadIdx.x & 31;
  if (wave >= M) return;
  long long base = (long long)wave * 64 + lane * 2;

  float a0 = nA[base], a1 = nA[base + 1];
  float b0 = nB[base], b1 = nB[base + 1];
  float c0 = nC[base], c1 = nC[base + 1];
  float d0 = nD[base], d1 = nD[base + 1];

  float ia = 1.f / fmaxf(sqrtf(waveSum(a0 * a0 + a1 * a1)), 1e-12f);
  float ib = 1.f / fmaxf(sqrtf(waveSum(b0 * b0 + b1 * b1)), 1e-12f);
  float ic = 1.f / fmaxf(sqrtf(waveSum(c0 * c0 + c1 * c1)), 1e-12f);
  float id = 1.f / fmaxf(sqrtf(waveSum(d0 * d0 + d1 * d1)), 1e-12f);

  const float third = 1.f / 3.f;
  out[base]     = (e0[base]     + e1[base]     + e2[base])     * third
                + cM * a0 * ia + cM * b0 * ib + cC * c0 * ic + cD * d0 * id;
  out[base + 1] = (e0[base + 1] + e1[base + 1] + e2[base + 1]) * third
                + cM * a1 * ia + cM * b1 * ib + cC * c1 * ic + cD * d1 * id;
}

// ---------------------------------------------------------------------------
extern "C" void kernel_launch(void* const* d_in, const int* in_sizes, int n_in,
                              void* d_out, int out_size, void* d_ws, size_t ws_size,
                              hipStream_t stream)
{
  const int*   ui_rows     = (const int*)d_in[0];
  const int*   ui_cols     = (const int*)d_in[1];
  const float* ui_vals     = (const float*)d_in[2];
  const float* image_feats = (const float*)d_in[3];
  const float* text_feats  = (const float*)d_in[4];
  const float* user_feats0 = (const float*)d_in[5];
  const float* item_title  = (const float*)d_in[6];
  const float* W_img = (const float*)d_in[7];
  const float* b_img = (const float*)d_in[8];
  const float* W_txt = (const float*)d_in[9];
  const float* b_txt = (const float*)d_in[10];
  const float* W_usr = (const float*)d_in[11];
  const float* b_usr = (const float*)d_in[12];
  const float* W_itm = (const float*)d_in[13];
  const float* b_itm = (const float*)d_in[14];
  const float* user_id_emb = (const float*)d_in[15];
  const float* item_id_emb = (const float*)d_in[16];

  float* out = (float*)d_out;
  float* ws  = (float*)d_ws;

  const int NU = 30000, NI = 20000, E = 64;
  const int NNZ = in_sizes[0];
  const int NMASK = 2000;

  // Output tuple layout (flat, return order)
  long long off = 0;
  float* u_g        = out + off; off += (long long)NU * E;
  float* i_g        = out + off; off += (long long)NI * E;
  float* image_item = out + off; off += (long long)NI * E;
  float* text_item  = out + off; off += (long long)NI * E;
  float* image_user = out + off; off += (long long)NU * E;
  float* text_user  = out + off; off += (long long)NU * E;
  float* usr_e      = out + off; off += (long long)NU * E;
  float* item_fo    = out + off; off += (long long)NI * E;
  float* user_prof  = out + off; off += (long long)NU * E;
  float* item_prof  = out + off; off += (long long)NI * E;
  float* user_ffi   = out + off; off += (long long)NU * E;
  float* mask_nodes = out + off; off += NMASK;
  float* raw_image  = out + off; off += (long long)NI * E;
  float* raw_text   = out + off; off += (long long)NI * E;
  float* raw_item   = out + off; off += (long long)NI * E;

  // Workspace intermediates (u1, i1, u2, i2)
  float* u1 = ws;
  float* i1 = u1 + (long long)NU * E;
  float* u2 = i1 + (long long)NI * E;
  float* i2 = u2 + (long long)NU * E;

  // Zero accumulation targets (whole d_out + used ws)
  zero_f32<<<(int)((off + 255) / 256), 256, 0, stream>>>(out, off);
  long long wsElems = (long long)(NU + NI + NU + NI) * E;
  zero_f32<<<(int)((wsElems + 255) / 256), 256, 0, stream>>>(ws, wsElems);

  // Dense projections (fp32 WMMA). Masked projections == raw (see analysis).
  gemm_f32_wmma<<<(NU + 127) / 128, 256, 0, stream>>>(user_feats0, W_usr, b_usr, usr_e,     NU, 1536);
  gemm_f32_wmma<<<(NI + 127) / 128, 256, 0, stream>>>(image_feats, W_img, b_img, raw_image, NI, 4096);
  gemm_f32_wmma<<<(NI + 127) / 128, 256, 0, stream>>>(text_feats,  W_txt, b_txt, raw_text,  NI, 768);
  gemm_f32_wmma<<<(NI + 127) / 128, 256, 0, stream>>>(item_title,  W_itm, b_itm, raw_item,  NI, 1536);

  const int spGrid = (int)(((long long)NNZ * 64 + 255) / 256);
  // spmm_ui: dst=rows (users), src=cols (items); spmm_iu: swapped
  spmm_atomic<<<spGrid, 256, 0, stream>>>(ui_rows, ui_cols, ui_vals, raw_image,  image_user, NNZ);
  spmm_atomic<<<spGrid, 256, 0, stream>>>(ui_cols, ui_rows, ui_vals, image_user, image_item, NNZ);
  spmm_atomic<<<spGrid, 256, 0, stream>>>(ui_rows, ui_cols, ui_vals, raw_text,   text_user,  NNZ);
  spmm_atomic<<<spGrid, 256, 0, stream>>>(ui_cols, ui_rows, ui_vals, text_user,  text_item,  NNZ);
  spmm_atomic<<<spGrid, 256, 0, stream>>>(ui_rows, ui_cols, ui_vals, raw_item,   user_ffi,   NNZ);
  spmm_atomic<<<spGrid, 256, 0, stream>>>(ui_cols, ui_rows, ui_vals, user_ffi,   item_fo,    NNZ);
  spmm_atomic<<<spGrid, 256, 0, stream>>>(ui_cols, ui_rows, ui_vals, usr_e,      item_prof,  NNZ);
  spmm_atomic<<<spGrid, 256, 0, stream>>>(ui_rows, ui_cols, ui_vals, item_prof,  user_prof,  NNZ);

  // ID-embedding propagation: u1=ui(i0); i1=iu(u1); u2=softmax(ui(i1)); i2=softmax(iu(u2))
  spmm_atomic<<<spGrid, 256, 0, stream>>>(ui_rows, ui_cols, ui_vals, item_id_emb, u1, NNZ);
  spmm_atomic<<<spGrid, 256, 0, stream>>>(ui_cols, ui_rows, ui_vals, u1, i1, NNZ);
  spmm_atomic<<<spGrid, 256, 0, stream>>>(ui_rows, ui_cols, ui_vals, i1, u2, NNZ);
  softmax64<<<(NU * 32 + 255) / 256, 256, 0, stream>>>(u2, NU);
  spmm_atomic<<<spGrid, 256, 0, stream>>>(ui_cols, ui_rows, ui_vals, u2, i2, NNZ);
  softmax64<<<(NI * 32 + 255) / 256, 256, 0, stream>>>(i2, NI);

  // Blend epilogues
  finalize_rows<<<(NU * 32 + 255) / 256, 256, 0, stream>>>(
      user_id_emb, u1, u2, image_user, text_user, user_prof, user_ffi,
      u_g, NU, 0.02f, 0.3f, 0.3f);
  finalize_rows<<<(NI * 32 + 255) / 256, 256, 0, stream>>>(
      item_id_emb, i1, i2, image_item, text_item, item_prof, item_fo,
      i_g, NI, 0.02f, 0.3f, 0.3f);

  fill_iota<<<(NMASK + 255) / 256, 256, 0, stream>>>(mask_nodes, NMASK);
}